// GeometryAwareProjector_21260088115898
// MI455X (gfx1250) — compile-verified
//
#include <hip/hip_runtime.h>

#define BEV   256
#define NPTS  (BEV * BEV)     // 65536
#define B_    4
#define K_    6
#define C_    128
#define HF_   32
#define WF_   88
#define HW_   (HF_ * WF_)     // 2816

typedef float v2f __attribute__((ext_vector_type(2)));
typedef float v8f __attribute__((ext_vector_type(8)));

// Fast reciprocal: single v_rcp_f32 (TRANS, co-executes with VALU)
__device__ __forceinline__ float fast_rcp(float x) {
  return __builtin_amdgcn_rcpf(x);
}

// ---------------------------------------------------------------------------
// Kernel 1: transpose features [B*K, C, H, W] -> NHWC [B*K, H*W, C]
// so that the 128 channels of one feature pixel are contiguous (512B taps).
// ---------------------------------------------------------------------------
__global__ __launch_bounds__(256) void feat_to_nhwc(const float* __restrict__ in,
                                                    float* __restrict__ out) {
  __shared__ float tile[32][33];
  const int bk  = blockIdx.z;              // 0..23
  const int hw0 = blockIdx.x * 32;         // 88 tiles over 2816
  const int c0  = blockIdx.y * 32;         // 4 tiles over 128
  const int tx  = threadIdx.x & 31;
  const int ty  = threadIdx.x >> 5;        // 0..7
  const float* src = in  + (size_t)bk * C_ * HW_;
  float*       dst = out + (size_t)bk * HW_ * C_;
#pragma unroll
  for (int r = 0; r < 4; ++r) {
    int c = c0 + ty + r * 8;
    tile[ty + r * 8][tx] = src[(size_t)c * HW_ + (hw0 + tx)];   // coalesced on hw
  }
  __syncthreads();
#pragma unroll
  for (int r = 0; r < 4; ++r) {
    int hw = hw0 + ty + r * 8;
    dst[(size_t)hw * C_ + (c0 + tx)] = tile[tx][ty + r * 8];    // coalesced on c
  }
}

// ---------------------------------------------------------------------------
// Kernel 2: per block: 128 consecutive BEV points of one batch.
// Phase A: WMMA projection (2x v_wmma_f32_16x16x4_f32 per wave = 16 points,
//          all 6 cameras) -> bilinear tap descriptors in LDS.
// Phase B: bandwidth-bound NHWC gather + weighted accumulate, coalesced writes.
// ---------------------------------------------------------------------------
__global__ __launch_bounds__(256) void bev_project(const float* __restrict__ nhwc,
                                                   const float* __restrict__ intr,
                                                   const float* __restrict__ extr,
                                                   float* __restrict__ out) {
  __shared__ float  Mlds[K_ * 12];        // M = intr @ extr[:3,:]  per cam: 3x4
  __shared__ int    baseL[128][K_];       // NHWC element index of tap00, -1 = invalid
  __shared__ int    packL[128][K_];       // (o01 elements) | (o10 elements << 16)
  __shared__ float4 wL[128][K_];          // bilinear weights w00,w01,w10,w11

  const int tid = threadIdx.x;
  const int b   = blockIdx.y;
  const int n0  = blockIdx.x * 128;

  // --- build the 6 projection matrices M[k] (3x4) for this batch ---
  if (tid < K_ * 12) {
    int k   = tid / 12;
    int rc  = tid % 12;
    int r   = rc >> 2;
    int col = rc & 3;
    const float* I = intr + (size_t)(b * K_ + k) * 9;
    const float* E = extr + (size_t)(b * K_ + k) * 16;
    Mlds[tid] = I[r * 3 + 0] * E[0 * 4 + col] +
                I[r * 3 + 1] * E[1 * 4 + col] +
                I[r * 3 + 2] * E[2 * 4 + col];
  }
  __syncthreads();

  // ------------------------- Phase A: WMMA projection -------------------------
  const int lane = tid & 31;
  const int wave = tid >> 5;          // 8 waves, 16 points each
  const int half = lane >> 4;         // 0: lanes 0-15, 1: lanes 16-31
  const int m    = lane & 15;         // A-matrix row / point index within wave
  const int pt   = wave * 16 + m;     // 0..127 point slot in block
  const int n    = n0 + pt;           // BEV flat index
  const int k0   = half * 2;          // A supplies K={0,1} or K={2,3}

  // A-matrix rows: WMMA1 rows = cam(0..3)*4 + {u,v,w,pad}; WMMA2 = cams {4,5,4,5}
  const int camA1 = m >> 2;
  const int camA2 = 4 + (camA1 & 1);
  const int rrow  = m & 3;

  v2f a1, a2, bv;
  if (rrow < 3) {
    a1[0] = Mlds[camA1 * 12 + rrow * 4 + k0];
    a1[1] = Mlds[camA1 * 12 + rrow * 4 + k0 + 1];
    a2[0] = Mlds[camA2 * 12 + rrow * 4 + k0];
    a2[1] = Mlds[camA2 * 12 + rrow * 4 + k0 + 1];
  } else {
    a1[0] = 0.0f; a1[1] = 0.0f;
    a2[0] = 0.0f; a2[1] = 0.0f;
  }

  // B-matrix: world points [x, y, 0, 1] for 16 points (N = lane%16)
  const float STEP = 102.4f / 255.0f;
  if (half == 0) {
    int j = n & (BEV - 1);
    int i = n >> 8;
    bv[0] = -51.2f + (float)j * STEP;   // K=0: x
    bv[1] = -51.2f + (float)i * STEP;   // K=1: y
  } else {
    bv[0] = 0.0f;                        // K=2: z
    bv[1] = 1.0f;                        // K=3: homogeneous 1
  }

  v8f z = {};
  // D = A x B : components (u,v,w per camera) x 16 points
  v8f d1 = __builtin_amdgcn_wmma_f32_16x16x4_f32(false, a1, false, bv, (short)0, z, false, false);
  v8f d2 = __builtin_amdgcn_wmma_f32_16x16x4_f32(false, a2, false, bv, (short)0, z, false, false);

  auto storeTap = [&](int cam, float uw, float vw, float ww) {
    float depth = fmaxf(ww, 1e-5f);
    float inv   = fast_rcp(depth) * 0.125f;   // sx = sy = 0.125
    float u = uw * inv;
    float v = vw * inv;
    bool valid = (ww > 1e-3f) && (u >= 0.0f) && (u <= (float)(WF_ - 1)) &&
                 (v >= 0.0f) && (v <= (float)(HF_ - 1));
    if (valid) {
      float x0 = floorf(u), y0 = floorf(v);
      float wx = u - x0,    wy = v - y0;
      int xi = (int)x0, yi = (int)y0;
      int o01 = (xi < WF_ - 1) ? C_ : 0;          // +x step, in elements
      int o10 = (yi < HF_ - 1) ? WF_ * C_ : 0;    // +y step, in elements
      baseL[pt][cam] = (((b * K_ + cam) * HF_ + yi) * WF_ + xi) * C_;
      packL[pt][cam] = o01 | (o10 << 16);
      wL[pt][cam] = make_float4((1.0f - wx) * (1.0f - wy), wx * (1.0f - wy),
                                (1.0f - wx) * wy,          wx * wy);
    } else {
      baseL[pt][cam] = -1;
    }
  };

  // lanes 0-15 own cams {0,1,4,5}; lanes 16-31 own cams {2,3}
  storeTap(2 * half,     d1[0], d1[1], d1[2]);
  storeTap(2 * half + 1, d1[4], d1[5], d1[6]);
  if (half == 0) {
    storeTap(4, d2[0], d2[1], d2[2]);
    storeTap(5, d2[4], d2[5], d2[6]);
  }
  __syncthreads();

  // ------------------------- Phase B: gather + accumulate -------------------------
  const int p  = tid & 127;            // point slot; consecutive across lanes -> coalesced out
  const int qh = tid >> 7;             // 0/1: which channel-quad half this pass
  const int nB = n0 + p;
  float* outB = out + (size_t)b * C_ * NPTS + nB;

  for (int it = 0; it < 16; ++it) {
    int q = it * 2 + qh;               // channel quad 0..31 (c = 4q .. 4q+3)
    float4 acc = make_float4(0.f, 0.f, 0.f, 0.f);
    int cnt = 0;
#pragma unroll
    for (int k = 0; k < K_; ++k) {
      int bidx = baseL[p][k];
      if (bidx >= 0) {
        ++cnt;
        int pk  = packL[p][k];
        int o01 = pk & 0xffff;
        int o10 = pk >> 16;
        const float* tap = nhwc + bidx + q * 4;
        float4 t00 = *(const float4*)(tap);
        float4 t01 = *(const float4*)(tap + o01);
        float4 t10 = *(const float4*)(tap + o10);
        float4 t11 = *(const float4*)(tap + o01 + o10);
        float4 w = wL[p][k];
        acc.x += w.x * t00.x + w.y * t01.x + w.z * t10.x + w.w * t11.x;
        acc.y += w.x * t00.y + w.y * t01.y + w.z * t10.y + w.w * t11.y;
        acc.z += w.x * t00.z + w.y * t01.z + w.z * t10.z + w.w * t11.z;
        acc.w += w.x * t00.w + w.y * t01.w + w.z * t10.w + w.w * t11.w;
      }
    }
    float s = (cnt > 0) ? fast_rcp((float)cnt) : 1.0f;
    int c = q * 4;
    outB[(size_t)(c + 0) * NPTS] = acc.x * s;
    outB[(size_t)(c + 1) * NPTS] = acc.y * s;
    outB[(size_t)(c + 2) * NPTS] = acc.z * s;
    outB[(size_t)(c + 3) * NPTS] = acc.w * s;
  }
}

// ---------------------------------------------------------------------------
extern "C" void kernel_launch(void* const* d_in, const int* in_sizes, int n_in,
                              void* d_out, int out_size, void* d_ws, size_t ws_size,
                              hipStream_t stream) {
  const float* features = (const float*)d_in[0];   // [B,K,C,HF,WF]
  const float* intr     = (const float*)d_in[1];   // [B,K,3,3]
  const float* extr     = (const float*)d_in[2];   // [B,K,4,4]
  float* out  = (float*)d_out;                     // [B,C,256,256]
  float* nhwc = (float*)d_ws;                      // [B,K,HF,WF,C] = 34.6 MB

  // Stage 1: channel transpose to NHWC (one-time 2x35 MB traffic, L2-resident after)
  feat_to_nhwc<<<dim3(HW_ / 32, C_ / 32, B_ * K_), 256, 0, stream>>>(features, nhwc);

  // Stage 2: WMMA projection + bilinear BEV splat
  bev_project<<<dim3(NPTS / 128, B_), 256, 0, stream>>>(nhwc, intr, extr, out);
}